// GraphConvolution_5403068858431
// MI455X (gfx1250) — compile-verified
//
#include <hip/hip_runtime.h>

typedef __attribute__((ext_vector_type(2))) float v2f;
typedef __attribute__((ext_vector_type(8))) float v8f;
typedef __attribute__((ext_vector_type(4))) unsigned int u32x4;
typedef __attribute__((ext_vector_type(8))) int i32x8;
typedef __attribute__((ext_vector_type(4))) int i32x4;

#define F_IN  256
#define F_HID 128

// ---------------------------------------------------------------------------
// Kernel 1: h = x @ w  via V_WMMA_F32_16X16X4_F32 (full fp32 precision).
// Block = 256 threads (8 waves). Block owns 16 rows; wave i owns cols [16i,16i+16).
// A tile (16x256 f32 = 16 KB, contiguous rows of x) staged into LDS by the
// Tensor Data Mover (TENSOR_LOAD_TO_LDS, TENSORcnt) issued by wave 0 only.
// B fragments read from global (w is 128 KB -> L2 resident across all blocks).
// ---------------------------------------------------------------------------
__global__ __launch_bounds__(256) void gemm_wmma_f32(
    const float* __restrict__ x, const float* __restrict__ w,
    float* __restrict__ h) {
  __shared__ float lds_a[16 * F_IN];  // 16 KB

  const int tid      = threadIdx.x;
  const int row_base = blockIdx.x * 16;

  // ---- TDM stage of the A tile (wave 0 issues; TDM ignores EXEC, so the
  // scalar branch keeps waves 1..7 from duplicating the DMA). ----
  if (tid < 32) {
    unsigned long long ga =
        (unsigned long long)(const void*)(x + (size_t)row_base * F_IN);
    unsigned ga_lo  = __builtin_amdgcn_readfirstlane((unsigned)ga);
    unsigned ga_hi  = __builtin_amdgcn_readfirstlane((unsigned)(ga >> 32));
    unsigned lds_ad = __builtin_amdgcn_readfirstlane((unsigned)(size_t)(void*)lds_a);

    // D# group 0: count=1 | lds_addr | global_addr[56:0] | type=2
    u32x4 g0;
    g0[0] = 1u;
    g0[1] = lds_ad;
    g0[2] = ga_lo;
    g0[3] = (ga_hi & 0x01FFFFFFu) | 0x80000000u;

    // D# group 1: data_size=2 (4B), tensor 256x16, tile 256x16, dim0 stride 256
    i32x8 g1;
    g1[0] = 0x20000;              // [17:16] data_size = 2 -> 4-byte elements
    g1[1] = (int)(256u << 16);    // tensor_dim0[15:0]=256 in bits [63:48]
    g1[2] = (int)(16u << 16);     // tensor_dim0 hi=0 | tensor_dim1[15:0]=16
    g1[3] = (int)(256u << 16);    // tensor_dim1 hi=0 | tile_dim0=256
    g1[4] = 16;                   // tile_dim1=16, tile_dim2=0
    g1[5] = 256;                  // tensor_dim0_stride[31:0]=256
    g1[6] = 0;
    g1[7] = 0;

    i32x4 gz4 = {0, 0, 0, 0};
    i32x8 gz8 = {0, 0, 0, 0, 0, 0, 0, 0};
    __builtin_amdgcn_tensor_load_to_lds(g0, g1, gz4, gz4, gz8, 0);
    __builtin_amdgcn_s_wait_tensorcnt(0);
  }
  __syncthreads();

  const int waveId   = tid >> 5;
  const int lane     = tid & 31;
  const int m        = lane & 15;          // A row / D row-within-half
  const int n        = lane & 15;          // B/D column
  const int koff     = (lane >> 4) << 1;   // lanes 16-31 hold K+2,K+3
  const int col_base = waveId * 16;

  v8f acc = {};
#pragma unroll 8
  for (int k0 = 0; k0 < F_IN; k0 += 4) {
    // A fragment: lane holds x[row_base+m][k0+koff .. +1]  (8B-aligned ds_load_b64)
    v2f a = *(const v2f*)&lds_a[m * F_IN + k0 + koff];
    // B fragment: lane holds w[k0+koff][col_base+n], w[k0+koff+1][col_base+n]
    v2f b;
    b.x = w[(k0 + koff    ) * F_HID + col_base + n];
    b.y = w[(k0 + koff + 1) * F_HID + col_base + n];
    // 8 args: (neg_a, A, neg_b, B, c_mod, C, reuse_a, reuse_b)
    acc = __builtin_amdgcn_wmma_f32_16x16x4_f32(
        false, a, false, b, (short)0, acc, false, false);
  }

  // D layout: VGPR r -> M = r (lanes 0-15) / r+8 (lanes 16-31), N = lane&15
  const int mrow = row_base + ((lane >> 4) << 3);
#pragma unroll
  for (int r = 0; r < 8; ++r) {
    h[(size_t)(mrow + r) * F_HID + col_base + n] = acc[r];
  }
}

// ---------------------------------------------------------------------------
// Kernel 2: out[r][:] = b[:]   (bias folded into the accumulator init)
// ---------------------------------------------------------------------------
__global__ __launch_bounds__(256) void init_bias(
    const float* __restrict__ b, float* __restrict__ out, int n_f4) {
  int i = blockIdx.x * 256 + threadIdx.x;     // float4 index over out
  if (i < n_f4) {
    float4 bv = ((const float4*)b)[i & 31];   // 128 floats = 32 float4 per row
    ((float4*)out)[i] = bv;
  }
}

// ---------------------------------------------------------------------------
// Kernel 3: COO scatter:  out[row[e]] += val[e] * h[col[e]]
// One wave per edge; lane owns a contiguous float4 slice of the 128 features.
// h and out both fit in the 192 MB L2 -> gathers + native f32 atomics resolve
// at L2 bandwidth (unsafeAtomicAdd == global_atomic_add_f32, no CAS loop).
// ---------------------------------------------------------------------------
__global__ __launch_bounds__(256) void spmm_scatter(
    const int* __restrict__ erow, const int* __restrict__ ecol,
    const float* __restrict__ eval, const float* __restrict__ h,
    float* __restrict__ out, int n_edges) {
  const int lane = threadIdx.x & 31;
  const int e    = blockIdx.x * 8 + (threadIdx.x >> 5);
  if (e >= n_edges) return;

  const int   r = erow[e];
  const int   c = ecol[e];
  const float v = eval[e];

  // gather: 128 floats of h[c], lane -> float4 slice (global_load_b128)
  float4 msg = ((const float4*)(h + (size_t)c * F_HID))[lane];

  float* o = out + (size_t)r * F_HID + lane * 4;
  unsafeAtomicAdd(o + 0, v * msg.x);
  unsafeAtomicAdd(o + 1, v * msg.y);
  unsafeAtomicAdd(o + 2, v * msg.z);
  unsafeAtomicAdd(o + 3, v * msg.w);
}

// ---------------------------------------------------------------------------
extern "C" void kernel_launch(void* const* d_in, const int* in_sizes, int n_in,
                              void* d_out, int out_size, void* d_ws, size_t ws_size,
                              hipStream_t stream) {
  const float* x    = (const float*)d_in[0];
  const int*   erow = (const int*)  d_in[1];
  const int*   ecol = (const int*)  d_in[2];
  const float* eval = (const float*)d_in[3];
  const float* w    = (const float*)d_in[4];
  const float* b    = (const float*)d_in[5];
  float*       out  = (float*)d_out;
  float*       h    = (float*)d_ws;            // [n_nodes, F_HID] scratch

  const int n_nodes = in_sizes[0] / F_IN;      // 100000
  const int n_edges = in_sizes[1];             // 1600000

  // 1) dense GEMM: one block per 16 rows (100000 = 6250 * 16 exactly)
  gemm_wmma_f32<<<n_nodes / 16, 256, 0, stream>>>(x, w, h);

  // 2) out = broadcast(bias)
  const int n_f4 = n_nodes * (F_HID / 4);
  init_bias<<<(n_f4 + 255) / 256, 256, 0, stream>>>(b, out, n_f4);

  // 3) edge scatter-add (8 edges per 256-thread block)
  spmm_scatter<<<(n_edges + 7) / 8, 256, 0, stream>>>(erow, ecol, eval, h, out, n_edges);
}